// set_attention_515396076025
// MI455X (gfx1250) — compile-verified
//
#include <hip/hip_runtime.h>

// CDNA5 / gfx1250 set-attention kernel.
// One workgroup (256 threads = 8 wave32) per batch element (1024 batches).
// All GEMMs run on V_WMMA_F32_16X16X4_F32 out of LDS.
// - B fragments: unconditional aligned ds_load_b64 (zero-padded, transposed LDS)
// - dual accumulators / multi-tile loops to break WMMA dependency chains
// - float4-vectorized masked softmax

typedef __attribute__((ext_vector_type(2))) float v2f;
typedef __attribute__((ext_vector_type(8))) float v8f;

#define HEAD_SIZE 20
#define NUM_HEADS 2
#define SEQ 256
#define NSET 32
#define DMODEL 20          // input feature dim
#define DQK 40             // HEAD_SIZE * NUM_HEADS
#define QKS 48             // padded row stride for Q/K in LDS
#define VTROWS 56          // padded row count for V^T in LDS

__device__ __forceinline__ v8f wmma4(v2f a, v2f b, v8f c) {
  // D = A(16x4) * B(4x16) + C(16x16), all f32
  return __builtin_amdgcn_wmma_f32_16x16x4_f32(
      /*neg_a=*/false, a, /*neg_b=*/false, b,
      /*c_mod=*/(short)0, c, /*reuse_a=*/false, /*reuse_b=*/false);
}

__global__ __launch_bounds__(256)
void set_attention_kernel(const float* __restrict__ x,
                          const float* __restrict__ y,
                          const float* __restrict__ WQg,
                          const float* __restrict__ WKg,
                          const float* __restrict__ WVg,
                          const float* __restrict__ WHg,
                          float* __restrict__ out) {
  extern __shared__ float smem[];
  // LDS layout (floats):
  float* Qs  = smem;                 // [256][48]  (cols 40..47 = exact zeros)
  float* Ks  = smem + 12288;         // [256][48]
  float* Vt  = smem + 24576;         // [56][256]  V^T, rows 40..55 zero
  float* Wqt = smem + 38912;         // [48][20]   WQ^T, rows 40..47 zero
  float* Wkt = Wqt + 960;            // [48][20]
  float* Wvt = Wkt + 960;            // [48][20]
  float* Wht = smem + 41792;         // [32][40]   WH^T, rows 20..31 zero
  float* Xs  = smem + 43072;         // [256][20]  (union with S strips)
  float* Ys  = Xs + 5120;            // [256][20]
  float* Sbase = smem + 43072;       // 8 strips of [16][256]
  // total = 43072 + 32768 = 75840 floats = 303360 bytes (< 320KB WGP LDS)

  const int tid  = threadIdx.x;
  const int wid  = tid >> 5;
  const int lane = tid & 31;
  const int ln   = lane & 15;  // M or N index inside a 16-wide tile
  const int lh   = lane >> 4;  // half-wave: selects K pair / row group

  // batch mapping per the reference's set-dim transposes:
  // flat batch b: a = b/32, c = b%32 ; q <- x[c,a], k,v <- y[a,c], out -> [c,a]
  const int b  = blockIdx.x;
  const int a  = b >> 5;
  const int cc = b & 31;
  const size_t xoff = (size_t)(cc * NSET + a) * (SEQ * DMODEL);
  const size_t yoff = (size_t)(a * NSET + cc) * (SEQ * DMODEL);
  const size_t ooff = (size_t)(cc * NSET + a) * (SEQ * HEAD_SIZE);

  // ---- stage x/y tiles (float4), weights (transposed), zero pads ----
  {
    const float4* gx = (const float4*)(x + xoff);
    const float4* gy = (const float4*)(y + yoff);
    float4* lx = (float4*)Xs;
    float4* ly = (float4*)Ys;
    for (int i = tid; i < (SEQ * DMODEL) / 4; i += 256) { lx[i] = gx[i]; ly[i] = gy[i]; }
    // W[20][40] -> Wt[n][k] (transposed); pad rows filled separately (disjoint)
    for (int i = tid; i < 800; i += 256) {
      const int k = i / DQK, n = i % DQK;
      Wqt[n * DMODEL + k] = WQg[i];
      Wkt[n * DMODEL + k] = WKg[i];
      Wvt[n * DMODEL + k] = WVg[i];
    }
    // WH[40][20] -> Wht[n][k]
    for (int i = tid; i < 800; i += 256) {
      const int k = i / HEAD_SIZE, n = i % HEAD_SIZE;
      Wht[n * DQK + k] = WHg[i];
    }
    // zero pad regions (disjoint from fills above -> no intra-phase race)
    for (int i = tid; i < 160; i += 256) {
      Wqt[800 + i] = 0.f; Wkt[800 + i] = 0.f; Wvt[800 + i] = 0.f;
    }
    for (int i = tid; i < 480; i += 256) Wht[800 + i] = 0.f;
    for (int i = tid; i < (VTROWS - QKS) * SEQ; i += 256) Vt[QKS * SEQ + i] = 0.f;
  }
  __syncthreads();

  // ---- projections: Q = X@WQ, K = Y@WK, V^T = (Y@WV)^T ----
  // Each wave: row blocks {wid, wid+8} x 3 col tiles; K/V share A fragments.
  for (int rbi = 0; rbi < 2; ++rbi) {
    const int rb   = wid + rbi * 8;
    const int arow = (rb * 16 + ln) * DMODEL;
    for (int ct = 0; ct < 3; ++ct) {
      const int col = ct * 16 + ln;
      v8f cq = {}, ck = {}, cv = {};
#pragma unroll
      for (int k0 = 0; k0 < DMODEL; k0 += 4) {
        const int kk = k0 + 2 * lh;
        v2f ax = *(const v2f*)&Xs[arow + kk];
        v2f ay = *(const v2f*)&Ys[arow + kk];
        cq = wmma4(ax, *(const v2f*)&Wqt[col * DMODEL + kk], cq);
        ck = wmma4(ay, *(const v2f*)&Wkt[col * DMODEL + kk], ck);
        cv = wmma4(ay, *(const v2f*)&Wvt[col * DMODEL + kk], cv);
      }
#pragma unroll
      for (int v = 0; v < 8; ++v) {
        const int row = rb * 16 + v + 8 * lh;
        Qs[row * QKS + col] = cq[v];
        Ks[row * QKS + col] = ck[v];
        Vt[col * SEQ + row] = cv[v];
      }
    }
  }
  __syncthreads();   // also fences X/Y staging region before S-strip reuse

  // ---- attention: each wave owns row blocks {wid, wid+8}, heads sequential ----
  const float scale = 0.22360679774997896f;  // 1/sqrt(20)
  float* Sw = Sbase + wid * (16 * SEQ);

  for (int h = 0; h < NUM_HEADS; ++h) {
    const int hoff = h * HEAD_SIZE;
    for (int rbi = 0; rbi < 2; ++rbi) {
      const int rb = wid + rbi * 8;

      // logits strip S[16][256] = Q_h[rb] @ K_h^T * scale (2 tiles at a time)
      {
        v2f aq[5];
#pragma unroll
        for (int i = 0; i < 5; ++i)
          aq[i] = *(const v2f*)&Qs[(rb * 16 + ln) * QKS + hoff + 4 * i + 2 * lh];
        for (int ct = 0; ct < 16; ct += 2) {
          v8f c0 = {}, c1 = {};
#pragma unroll
          for (int i = 0; i < 5; ++i) {
            const int kk = hoff + 4 * i + 2 * lh;
            c0 = wmma4(aq[i], *(const v2f*)&Ks[(ct * 16 + ln) * QKS + kk], c0);
            c1 = wmma4(aq[i], *(const v2f*)&Ks[((ct + 1) * 16 + ln) * QKS + kk], c1);
          }
#pragma unroll
          for (int v = 0; v < 8; ++v) {
            Sw[(v + 8 * lh) * SEQ + ct * 16 + ln]       = c0[v] * scale;
            Sw[(v + 8 * lh) * SEQ + (ct + 1) * 16 + ln] = c1[v] * scale;
          }
        }
      }

      // masked softmax: lane pair (ln, ln+16) handles row ln, 128 cols per half
      {
        float4* row = (float4*)&Sw[ln * SEQ + lh * 128];   // 256B aligned
        float mx = -__builtin_inff();
        for (int i = 0; i < 32; ++i) {
          const float4 r = row[i];
          mx = fmaxf(mx, fmaxf(fmaxf(r.x, r.y), fmaxf(r.z, r.w)));
        }
        mx = fmaxf(mx, __shfl_xor(mx, 16, 32));
        float sum = 0.f;
        for (int i = 0; i < 32; ++i) {
          const float4 r = row[i];
          float4 e;
          e.x = (r.x != 0.f) ? __expf(r.x - mx) : 0.f;   // zero-logit mask
          e.y = (r.y != 0.f) ? __expf(r.y - mx) : 0.f;
          e.z = (r.z != 0.f) ? __expf(r.z - mx) : 0.f;
          e.w = (r.w != 0.f) ? __expf(r.w - mx) : 0.f;
          row[i] = e;
          sum += (e.x + e.y) + (e.z + e.w);
        }
        sum += __shfl_xor(sum, 16, 32);
        const float inv = 1.f / (sum + 1e-10f);
        for (int i = 0; i < 32; ++i) {
          float4 r = row[i];
          r.x *= inv; r.y *= inv; r.z *= inv; r.w *= inv;
          row[i] = r;
        }
      }

      // O = P @ V_h ; B = single b64 from V^T (rows >= 40 zero); dual accum
      for (int ct = 0; ct < 2; ++ct) {
        const int col  = ct * 16 + ln;
        const int vrow = hoff + col;           // <= 51 < VTROWS
        v8f c0 = {}, c1 = {};
        for (int k0 = 0; k0 < SEQ; k0 += 8) {
          const int kk = k0 + 2 * lh;
          c0 = wmma4(*(const v2f*)&Sw[ln * SEQ + kk],
                     *(const v2f*)&Vt[vrow * SEQ + kk], c0);
          c1 = wmma4(*(const v2f*)&Sw[ln * SEQ + kk + 4],
                     *(const v2f*)&Vt[vrow * SEQ + kk + 4], c1);
        }
        const v8f c = c0 + c1;
        if (col < HEAD_SIZE) {                 // per-column guard: only cols 0..19 stored
#pragma unroll
          for (int v = 0; v < 8; ++v)
            Qs[(rb * 16 + v + 8 * lh) * QKS + hoff + col] = c[v];  // O in place
        }
      }
    }
  }

  // ---- output projection: out = O @ WH (rows exclusively owned per wave) ----
  for (int rbi = 0; rbi < 2; ++rbi) {
    const int rb = wid + rbi * 8;
    for (int ct = 0; ct < 2; ++ct) {
      const int col = ct * 16 + ln;            // < 32; Wht rows 20..31 zero
      v8f c0 = {}, c1 = {};
#pragma unroll
      for (int k0 = 0; k0 < DQK; k0 += 8) {
        const int kk = k0 + 2 * lh;
        c0 = wmma4(*(const v2f*)&Qs[(rb * 16 + ln) * QKS + kk],
                   *(const v2f*)&Wht[col * DQK + kk], c0);
        c1 = wmma4(*(const v2f*)&Qs[(rb * 16 + ln) * QKS + kk + 4],
                   *(const v2f*)&Wht[col * DQK + kk + 4], c1);
      }
      const v8f c = c0 + c1;
      if (col < HEAD_SIZE) {
        float* og = out + ooff;
#pragma unroll
        for (int v = 0; v < 8; ++v)
          og[(rb * 16 + v + 8 * lh) * HEAD_SIZE + col] = c[v];
      }
    }
  }
}

extern "C" void kernel_launch(void* const* d_in, const int* in_sizes, int n_in,
                              void* d_out, int out_size, void* d_ws, size_t ws_size,
                              hipStream_t stream) {
  (void)in_sizes; (void)n_in; (void)out_size; (void)d_ws; (void)ws_size;
  const float* x  = (const float*)d_in[0];
  const float* y  = (const float*)d_in[1];
  const float* WQ = (const float*)d_in[2];
  const float* WK = (const float*)d_in[3];
  const float* WV = (const float*)d_in[4];
  const float* WH = (const float*)d_in[5];
  float* out = (float*)d_out;

  const size_t shmem = (size_t)75840 * sizeof(float);  // 303360 B dynamic LDS
  (void)hipFuncSetAttribute((const void*)set_attention_kernel,
                            hipFuncAttributeMaxDynamicSharedMemorySize,
                            (int)shmem);
  hipLaunchKernelGGL(set_attention_kernel,
                     dim3(NSET * NSET), dim3(256), shmem, stream,
                     x, y, WQ, WK, WV, WH, out);
}